// GridSampleCrossBEVAttention_36919538876454
// MI455X (gfx1250) — compile-verified
//
#include <hip/hip_runtime.h>

typedef __attribute__((ext_vector_type(2))) float v2f;
typedef __attribute__((ext_vector_type(8))) float v8f;
typedef __attribute__((ext_vector_type(4))) unsigned int u32x4;
typedef __attribute__((ext_vector_type(8))) int i32x8;
typedef __attribute__((ext_vector_type(4))) int i32x4;

#define BS   16
#define NQ   1024
#define NP   8
#define ND   256
#define CIN  64
#define CH   128
#define CW   128
#define COUT 256

// ---------------------------------------------------------------------------
// Pack conv_w (OIHW) into WMMA-native Wt4[kblk][n][4] with K-order
// k = tap*64 + ci (tap-major, ci-inner), kblk = k/4.  144*256*4 floats.
// ---------------------------------------------------------------------------
__global__ __launch_bounds__(256)
void pack_convw_kernel(const float* __restrict__ W, float* __restrict__ Wt4)
{
    const int idx  = blockIdx.x * 256 + threadIdx.x;  // kblk*256 + n
    const int kblk = idx >> 8;
    const int n    = idx & 255;
    float tmp[4];
#pragma unroll
    for (int j = 0; j < 4; ++j) {
        int k   = (kblk << 2) + j;
        int ci  = k & 63;
        int tap = k >> 6;
        tmp[j] = W[n * 576 + ci * 9 + tap];
    }
    float4* dst = (float4*)(Wt4 + ((size_t)idx << 2));
    *dst = make_float4(tmp[0], tmp[1], tmp[2], tmp[3]);
}

// Pack out_w[n][k] into OWt4[kblk][n][4] (k natural order). 64*256*4 floats.
__global__ __launch_bounds__(256)
void pack_outw_kernel(const float* __restrict__ OW, float* __restrict__ OWt4)
{
    const int idx  = blockIdx.x * 256 + threadIdx.x;
    const int kblk = idx >> 8;
    const int n    = idx & 255;
    float tmp[4];
#pragma unroll
    for (int j = 0; j < 4; ++j)
        tmp[j] = OW[n * 256 + (kblk << 2) + j];
    float4* dst = (float4*)(OWt4 + ((size_t)idx << 2));
    *dst = make_float4(tmp[0], tmp[1], tmp[2], tmp[3]);
}

// ---------------------------------------------------------------------------
// Kernel 1: attn = softmax(queries @ attn_w^T + attn_b) over P=8
// ---------------------------------------------------------------------------
__global__ __launch_bounds__(256)
void attn_kernel(const float* __restrict__ Q, const float* __restrict__ AW,
                 const float* __restrict__ AB, float* __restrict__ attn)
{
    const int wave = threadIdx.x >> 5;
    const int lane = threadIdx.x & 31;
    const int qi   = blockIdx.x * 8 + wave;
    const float* qp = Q + (size_t)qi * ND;

    float part[NP];
#pragma unroll
    for (int p = 0; p < NP; ++p) part[p] = 0.f;

    for (int d = lane; d < ND; d += 32) {
        float qv = qp[d];
#pragma unroll
        for (int p = 0; p < NP; ++p) part[p] += qv * AW[p * ND + d];
    }
#pragma unroll
    for (int p = 0; p < NP; ++p) {
#pragma unroll
        for (int off = 16; off >= 1; off >>= 1)
            part[p] += __shfl_xor(part[p], off, 32);
        part[p] += AB[p];
    }
    float m = part[0];
#pragma unroll
    for (int p = 1; p < NP; ++p) m = fmaxf(m, part[p]);
    float s = 0.f;
#pragma unroll
    for (int p = 0; p < NP; ++p) { part[p] = __expf(part[p] - m); s += part[p]; }
    float inv = 1.f / s;
    if (lane == 0) {
#pragma unroll
        for (int p = 0; p < NP; ++p)
            attn[(size_t)qi * NP + p] = part[p] * inv;
    }
}

// ---------------------------------------------------------------------------
// Kernel 2: value = relu(conv3x3(bev) + conv_b), NHWC out, implicit GEMM
// via V_WMMA_F32_16X16X4_F32.  B fragments come from packed Wt4 as single
// aligned b64 loads; A fragments from the LDS input patch.
// ---------------------------------------------------------------------------
__global__ __launch_bounds__(512)
void conv_wmma_kernel(const float* __restrict__ X, const float* __restrict__ Wt4,
                      const float* __restrict__ bias, float* __restrict__ Yn)
{
    __shared__ __align__(16) float patch[CIN * 54];   // [ci][kh(3)][col(18)]
    const int bid = blockIdx.x;
    const int xt  = bid & 7;
    const int y   = (bid >> 3) & 127;
    const int b   = bid >> 10;
    const int tid = threadIdx.x;

    for (int e = tid; e < CIN * 54; e += 512) {
        int ci = e / 54, r = e % 54;
        int kh = r / 18, col = r % 18;
        int iy = y + kh - 1;
        int ix = (xt << 4) + col - 1;
        float v = 0.f;
        if ((unsigned)iy < CH && (unsigned)ix < CW)
            v = X[(((size_t)b * CIN + ci) * CH + iy) * CW + ix];
        patch[e] = v;
    }
    __syncthreads();

    const int wave = tid >> 5;
    const int lane = tid & 31;
    const int n    = (wave << 4) + (lane & 15);
    const int m    = lane & 15;
    const int hi   = lane >> 4;

    float bv = bias[n];
    v8f acc = { bv, bv, bv, bv, bv, bv, bv, bv };

#pragma unroll
    for (int tap = 0; tap < 9; ++tap) {
        const int kh = tap / 3, kw = tap % 3;
        const float* prow = patch + kh * 18 + m + kw;
        const float* wtap = Wt4 + (((size_t)(tap * 16) * 256 + n) << 2) + (hi << 1);
        if (tap < 8)   // prefetch next tap's weight stream (global_prefetch_b8)
            __builtin_prefetch(Wt4 + (((size_t)((tap + 1) * 16) * 256 + n) << 2), 0, 1);
#pragma unroll 4
        for (int cb = 0; cb < 16; ++cb) {
            int c0 = (cb << 2) + (hi << 1);
            v2f a, bb;
            a.x = prow[c0 * 54];
            a.y = prow[(c0 + 1) * 54];
            bb  = *(const v2f*)(wtap + ((size_t)cb << 10));   // kblk stride = 256*4
            acc = __builtin_amdgcn_wmma_f32_16x16x4_f32(
                false, a, false, bb, (short)0, acc, false, false);
        }
    }
    const int px = (xt << 4);
#pragma unroll
    for (int r = 0; r < 8; ++r) {
        int M = r + (hi << 3);
        float v = acc[r] > 0.f ? acc[r] : 0.f;
        Yn[(((size_t)b * CH + y) * CW + (px + M)) * COUT + n] = v;
    }
}

// ---------------------------------------------------------------------------
// Kernel 3: bilinear grid sample + attention-weighted reduce over P
// ---------------------------------------------------------------------------
__global__ __launch_bounds__(256)
void sample_kernel(const float* __restrict__ traj, const float* __restrict__ attn,
                   const float* __restrict__ Yn, float* __restrict__ mid)
{
    __shared__ int   sIdx[NP * 4];
    __shared__ float sWgt[NP * 4];
    const int bq  = blockIdx.x;
    const int b   = bq >> 10;
    const int tid = threadIdx.x;

    if (tid < NP) {
        const int p = tid;
        float t0 = traj[((size_t)bq * NP + p) * 2 + 0];
        float t1 = traj[((size_t)bq * NP + p) * 2 + 1];
        float gx = t1 * (1.0f / 32.0f);
        float gy = t0 * (1.0f / 32.0f);
        float x = (gx + 1.0f) * (CW * 0.5f) - 0.5f;
        float y = (gy + 1.0f) * (CH * 0.5f) - 0.5f;
        float x0f = floorf(x), y0f = floorf(y);
        int x0 = (int)x0f, y0 = (int)y0f;
        float wx1 = x - x0f, wx0 = 1.f - wx1;
        float wy1 = y - y0f, wy0 = 1.f - wy1;
        float ap = attn[(size_t)bq * NP + p];
        const int   xs[2]  = { x0, x0 + 1 };
        const int   ys[2]  = { y0, y0 + 1 };
        const float wxs[2] = { wx0, wx1 };
        const float wys[2] = { wy0, wy1 };
#pragma unroll
        for (int j = 0; j < 4; ++j) {
            int ix = xs[j & 1], iy = ys[j >> 1];
            bool ok = ((unsigned)ix < CW) && ((unsigned)iy < CH);
            sIdx[p * 4 + j] = ok ? (iy * CW + ix) : 0;
            sWgt[p * 4 + j] = ok ? (ap * wxs[j & 1] * wys[j >> 1]) : 0.f;
        }
    }
    __syncthreads();

    const int c = tid;
    const float* base = Yn + (size_t)b * CH * CW * COUT + c;
    float acc = 0.f;
#pragma unroll 4
    for (int j = 0; j < NP * 4; ++j)
        acc += sWgt[j] * base[(size_t)sIdx[j] * COUT];
    mid[(size_t)bq * COUT + c] = acc;
}

// ---------------------------------------------------------------------------
// Kernel 4: out = mid @ out_w^T + out_b + queries  (M=16384, N=256, K=256)
// A-tile (16x256 f32 = 16KB contiguous) loaded into LDS by the Tensor Data
// Mover (one tensor_load_to_lds from wave 0; TDM ignores EXEC), synced with
// s_wait_tensorcnt + workgroup barrier.  B from packed OWt4.
// ---------------------------------------------------------------------------
__global__ __launch_bounds__(512)
void proj_wmma_kernel(const float* __restrict__ mid, const float* __restrict__ OWt4,
                      const float* __restrict__ OB, const float* __restrict__ Q,
                      float* __restrict__ out)
{
    __shared__ __align__(16) float At[16 * ND];   // 16 KB
    const int m0   = blockIdx.x << 4;
    const int tid  = threadIdx.x;
    const int wave = tid >> 5;
    const int lane = tid & 31;

    if (wave == 0) {
        // Build Tensor DMA Descriptor (D#): 1D tile of 4096 f32 elements.
        unsigned long long gaddr = (unsigned long long)(const void*)(mid + (size_t)m0 * ND);
        unsigned int lds = (unsigned int)(size_t)(void*)At;   // LDS byte offset
        u32x4 g0;
        g0[0] = 1u;                                  // count=1 (valid), user flags 0
        g0[1] = lds;                                 // lds_addr [63:32]
        g0[2] = (unsigned int)gaddr;                 // global_addr [95:64]
        g0[3] = (unsigned int)((gaddr >> 32) & 0x1FFFFFFu) | (2u << 30); // addr hi + type=2
        i32x8 g1;
        g1[0] = 0x00020000;        // data_size=2 (4 bytes), no multicast/flags
        g1[1] = 0;                 // tensor_dim0[15:0] = 0  (dim0 = 0x400000)
        g1[2] = 0x00010040;        // tensor_dim0[31:16]=0x40, tensor_dim1[15:0]=1
        g1[3] = (int)(4096u << 16);// tile_dim0 = 4096
        g1[4] = 1;                 // tile_dim1 = 1, tile_dim2 = 0
        g1[5] = 0x400000;          // tensor_dim0_stride (low 32)
        g1[6] = 0;
        g1[7] = 0;
        i32x4 z4 = { 0, 0, 0, 0 };
        i32x8 z8 = { 0, 0, 0, 0, 0, 0, 0, 0 };
        __builtin_amdgcn_tensor_load_to_lds(g0, g1, z4, z4, z8, 0);
        __builtin_amdgcn_s_wait_tensorcnt(0);
    }
    __syncthreads();

    const int n  = (wave << 4) + (lane & 15);
    const int m  = lane & 15;
    const int hi = lane >> 4;

    v8f acc = { 0.f, 0.f, 0.f, 0.f, 0.f, 0.f, 0.f, 0.f };
    const float* wbase = OWt4 + ((size_t)n << 2) + (hi << 1);
    for (int kb = 0; kb < 64; ++kb) {
        int kk = (kb << 2) + (hi << 1);
        v2f a  = *(const v2f*)(&At[m * ND + kk]);                 // ds_load_b64
        v2f bb = *(const v2f*)(wbase + ((size_t)kb << 10));       // global b64
        acc = __builtin_amdgcn_wmma_f32_16x16x4_f32(
            false, a, false, bb, (short)0, acc, false, false);
    }
    float bn = OB[n];
#pragma unroll
    for (int r = 0; r < 8; ++r) {
        int M = r + (hi << 3);
        size_t o = ((size_t)(m0 + M)) * ND + n;
        out[o] = acc[r] + bn + Q[o];
    }
}

// ---------------------------------------------------------------------------
extern "C" void kernel_launch(void* const* d_in, const int* in_sizes, int n_in,
                              void* d_out, int out_size, void* d_ws, size_t ws_size,
                              hipStream_t stream)
{
    const float* queries = (const float*)d_in[0];  // (16,1024,256)
    const float* traj    = (const float*)d_in[1];  // (16,1024,8,2)
    const float* bev     = (const float*)d_in[2];  // (16,64,128,128)
    const float* attn_w  = (const float*)d_in[4];  // (8,256)
    const float* attn_b  = (const float*)d_in[5];  // (8,)
    const float* conv_w  = (const float*)d_in[6];  // (256,64,3,3)
    const float* conv_b  = (const float*)d_in[7];  // (256,)
    const float* out_w   = (const float*)d_in[8];  // (256,256)
    const float* out_b   = (const float*)d_in[9];  // (256,)
    float* out = (float*)d_out;                    // (16,1024,256)

    // workspace partition (all offsets 256B-aligned)
    char* ws = (char*)d_ws;
    float* attn_buf = (float*)(ws + 0);                       // 512 KB
    float* Wt4      = (float*)(ws + 524288);                  // 576 KB packed conv_w
    float* OWt4     = (float*)(ws + 524288 + 589824);         // 256 KB packed out_w
    float* value    = (float*)(ws + 524288 + 589824 + 262144);            // 256 MB NHWC
    float* mid      = (float*)(ws + 524288 + 589824 + 262144
                               + (size_t)BS * CH * CW * COUT * 4);        // 16 MB

    pack_convw_kernel<<<144, 256, 0, stream>>>(conv_w, Wt4);
    pack_outw_kernel<<<64, 256, 0, stream>>>(out_w, OWt4);
    attn_kernel<<<(BS * NQ) / 8, 256, 0, stream>>>(queries, attn_w, attn_b, attn_buf);
    conv_wmma_kernel<<<BS * CH * (CW / 16), 512, 0, stream>>>(bev, Wt4, conv_b, value);
    sample_kernel<<<BS * NQ, 256, 0, stream>>>(traj, attn_buf, value, mid);
    proj_wmma_kernel<<<(BS * NQ) / 16, 512, 0, stream>>>(mid, OWt4, out_b, queries, out);
}